// GLQuasiAlphaAttention_63934883168449
// MI455X (gfx1250) — compile-verified
//
#include <hip/hip_runtime.h>
#include <math.h>

// ---------------------------------------------------------------------------
// Problem constants (from the reference)
// ---------------------------------------------------------------------------
#define LL   12      // L_LAYERS
#define TT   8       // T
#define BB   32      // B
#define CC   256     // C
#define SS   64      // S
#define EE   64      // E
#define HDD  16      // head dim
#define NHH  4       // attention heads
#define HIDD 32      // MLP hidden
#define HH   8       // LU heads
#define NKK  8       // NK
#define MROWS (BB*CC*LL)        // 98304 rows of the (B*C, L, E) flat buffer
#define SROWS (LL*TT*BB*CC)     // 786432 stats rows

typedef float v2f __attribute__((ext_vector_type(2)));
typedef float v8f __attribute__((ext_vector_type(8)));

// native f32 WMMA: D(16x16) = A(16x4) * B(4x16) + C   -> v_wmma_f32_16x16x4_f32
static __device__ __forceinline__ v8f wmma4(v2f a, v2f b, v8f c) {
  return __builtin_amdgcn_wmma_f32_16x16x4_f32(false, a, false, b, (short)0, c,
                                               false, false);
}

static __device__ __forceinline__ float softplusf(float x) {
  return (x > 20.0f) ? x : log1pf(__expf(x));
}

// ---------------------------------------------------------------------------
// K1: per-row stats (mean, var ddof=1, quantiles 0/.25/.5/.75/1) + transpose
//     one wave32 per 64-element row; bitonic sort with 2 regs/lane.
//     feats layout: (B, L, C, 56), feats[b,l,c,t*7+s] = stats[l,t,b,c,s]
// ---------------------------------------------------------------------------
__global__ __launch_bounds__(256) void k_stats(const float* __restrict__ ws,
                                               float* __restrict__ feats) {
  const int lane = threadIdx.x & 31;
  const int wid  = threadIdx.x >> 5;
  const int row  = blockIdx.x * 8 + wid;              // grid = SROWS/8 exactly

  const float* p = ws + (size_t)row * SS;
  float v0 = p[lane];
  float v1 = p[lane + 32];

  // mean / var(ddof=1)
  float s = v0 + v1, s2 = v0 * v0 + v1 * v1;
  #pragma unroll
  for (int m = 16; m >= 1; m >>= 1) {
    s  += __shfl_xor(s,  m, 32);
    s2 += __shfl_xor(s2, m, 32);
  }
  const float mean = s * (1.0f / 64.0f);
  const float var  = (s2 - 64.0f * mean * mean) * (1.0f / 63.0f);

  // bitonic sort, element index i = lane + 32*r, ascending
  #pragma unroll
  for (int k = 2; k <= 64; k <<= 1) {
    #pragma unroll
    for (int j = k >> 1; j >= 1; j >>= 1) {
      if (j == 32) {                 // only at k==64, always ascending
        float lo = fminf(v0, v1), hi = fmaxf(v0, v1);
        v0 = lo; v1 = hi;
      } else {
        const bool lower = ((lane & j) == 0);
        {
          float o = __shfl_xor(v0, j, 32);
          bool asc = (k >= 64) ? true : ((lane & k) == 0);   // i=lane
          v0 = (asc == lower) ? fminf(v0, o) : fmaxf(v0, o);
        }
        {
          float o = __shfl_xor(v1, j, 32);
          bool asc;                                           // i=lane+32
          if (k >= 64)      asc = true;
          else if (k == 32) asc = false;
          else              asc = ((lane & k) == 0);
          v1 = (asc == lower) ? fminf(v1, o) : fmaxf(v1, o);
        }
      }
    }
  }

  // order statistics for linear-interp quantiles at 0, 15.75, 31.5, 47.25, 63
  const float q0  = __shfl(v0, 0, 32);
  const float s15 = __shfl(v0, 15, 32), s16 = __shfl(v0, 16, 32);
  const float s31 = __shfl(v0, 31, 32), s32 = __shfl(v1, 0, 32);
  const float s47 = __shfl(v1, 15, 32), s48 = __shfl(v1, 16, 32);
  const float q4  = __shfl(v1, 31, 32);
  const float q1 = s15 + 0.75f * (s16 - s15);
  const float q2 = 0.5f * (s31 + s32);
  const float q3 = s47 + 0.25f * (s48 - s47);

  if (lane == 0) {
    const int c = row & 255;
    int tmp = row >> 8;
    const int b = tmp & 31; tmp >>= 5;
    const int t = tmp & 7;
    const int l = tmp >> 3;
    float* o = feats + ((size_t)((b * LL + l) * CC + c)) * 56 + t * 7;
    o[0] = mean; o[1] = var; o[2] = q0; o[3] = q1; o[4] = q2; o[5] = q3; o[6] = q4;
  }
}

// ---------------------------------------------------------------------------
// K2: emb = feats(98304x56) @ W_emb(56x64) + b_emb + pos[l]   (WMMA f32)
//     block = 8 waves: 32 rows x 64 cols per block
// ---------------------------------------------------------------------------
__global__ __launch_bounds__(256) void k_embed(const float* __restrict__ feats,
                                               const float* __restrict__ Wemb,
                                               const float* __restrict__ bemb,
                                               const float* __restrict__ pos,
                                               float* __restrict__ emb) {
  const int lane = threadIdx.x & 31;
  const int wid  = threadIdx.x >> 5;
  const int r  = lane & 15;
  const int hf = lane >> 4;
  const int m0 = blockIdx.x * 32 + (wid >> 2) * 16;
  const int n0 = (wid & 3) * 16;

  v8f acc = {0, 0, 0, 0, 0, 0, 0, 0};
  #pragma unroll
  for (int k0 = 0; k0 < 56; k0 += 4) {
    const int c0 = k0 + 2 * hf;
    const float* ap = feats + (size_t)(m0 + r) * 56 + c0;
    v2f a; a.x = ap[0]; a.y = ap[1];
    const float* bp = Wemb + (size_t)c0 * EE + n0 + r;
    v2f b; b.x = bp[0]; b.y = bp[EE];
    acc = wmma4(a, b, acc);
  }
  #pragma unroll
  for (int g = 0; g < 8; ++g) {
    const int m = m0 + g + 8 * hf;
    const int n = n0 + r;
    const int l = (m >> 8) % LL;           // m = (b*12+l)*256 + c
    emb[(size_t)m * EE + n] = acc[g] + bemb[n] + pos[l * EE + n];
  }
}

// ---------------------------------------------------------------------------
// K3: qkv = emb(98304x64) @ w_in(64x192) + b_in   (WMMA f32)
//     block = 12 waves, one 16-col tile per wave, 16 shared rows per block
// ---------------------------------------------------------------------------
__global__ __launch_bounds__(384) void k_qkv(const float* __restrict__ emb,
                                             const float* __restrict__ win,
                                             const float* __restrict__ bin,
                                             float* __restrict__ qkv) {
  const int lane = threadIdx.x & 31;
  const int wid  = threadIdx.x >> 5;      // 0..11
  const int r  = lane & 15;
  const int hf = lane >> 4;
  const int m0 = blockIdx.x * 16;
  const int n0 = wid * 16;

  v8f acc = {0, 0, 0, 0, 0, 0, 0, 0};
  #pragma unroll
  for (int k0 = 0; k0 < 64; k0 += 4) {
    const int c0 = k0 + 2 * hf;
    const float* ap = emb + (size_t)(m0 + r) * EE + c0;
    v2f a; a.x = ap[0]; a.y = ap[1];
    const float* bp = win + (size_t)c0 * 192 + n0 + r;
    v2f b; b.x = bp[0]; b.y = bp[192];
    acc = wmma4(a, b, acc);
  }
  #pragma unroll
  for (int g = 0; g < 8; ++g) {
    const int m = m0 + g + 8 * hf;
    const int n = n0 + r;
    qkv[(size_t)m * 192 + n] = acc[g] + bin[n];
  }
}

// ---------------------------------------------------------------------------
// K4: attention. One wave per (batch n, head h): S = qk^T/4, softmax, P@v.
//     P goes C-layout -> LDS -> A-layout for the second WMMA chain.
// ---------------------------------------------------------------------------
__global__ __launch_bounds__(256) void k_attn(const float* __restrict__ qkv,
                                              float* __restrict__ ctx) {
  __shared__ float sP[8][16][17];
  const int lane = threadIdx.x & 31;
  const int wid  = threadIdx.x >> 5;
  const int task = blockIdx.x * 8 + wid;       // 32768 tasks exactly
  const int nb = task >> 2;
  const int h  = task & 3;
  const int r  = lane & 15;
  const int hf = lane >> 4;
  const float* base = qkv + (size_t)nb * LL * 192 + h * HDD;

  // scores S(16x16, valid 12x12) = q @ k^T, K=16
  v8f S = {0, 0, 0, 0, 0, 0, 0, 0};
  #pragma unroll
  for (int k0 = 0; k0 < 16; k0 += 4) {
    const int c0 = k0 + 2 * hf;
    v2f a = {0, 0}, b = {0, 0};
    if (r < LL) {
      const float* qp = base + (size_t)r * 192 + c0;
      a.x = qp[0]; a.y = qp[1];
      const float* kp = qp + EE;
      b.x = kp[0]; b.y = kp[1];
    }
    S = wmma4(a, b, S);
  }

  // softmax over n (=lane%16 group of 16 lanes), cols >=12 masked
  float P[8];
  #pragma unroll
  for (int g = 0; g < 8; ++g) {
    float sv = (r < LL) ? S[g] * 0.25f : -__builtin_inff();
    float rm = sv;
    #pragma unroll
    for (int m = 8; m >= 1; m >>= 1) rm = fmaxf(rm, __shfl_xor(rm, m, 32));
    float e = __expf(sv - rm);
    float rs = e;
    #pragma unroll
    for (int m = 8; m >= 1; m >>= 1) rs += __shfl_xor(rs, m, 32);
    P[g] = e / rs;
  }
  #pragma unroll
  for (int g = 0; g < 8; ++g) sP[wid][g + 8 * hf][r] = P[g];
  __syncthreads();

  // ctx(16x16, valid 12x16) = P @ v, K=16 (P cols >=12 are exactly 0)
  v8f Cv = {0, 0, 0, 0, 0, 0, 0, 0};
  #pragma unroll
  for (int k0 = 0; k0 < 16; k0 += 4) {
    const int c0 = k0 + 2 * hf;
    v2f a; a.x = sP[wid][r][c0]; a.y = sP[wid][r][c0 + 1];
    v2f b;
    b.x = (c0     < LL) ? base[(size_t)c0       * 192 + 128 + r] : 0.0f;
    b.y = (c0 + 1 < LL) ? base[(size_t)(c0 + 1) * 192 + 128 + r] : 0.0f;
    Cv = wmma4(a, b, Cv);
  }
  #pragma unroll
  for (int g = 0; g < 8; ++g) {
    const int m = g + 8 * hf;
    if (m < LL)
      ctx[((size_t)nb * LL + m) * EE + h * HDD + r] = Cv[g];
  }
}

// ---------------------------------------------------------------------------
// K5: attended = ctx @ w_out + b_out, then mean over each contiguous group of
//     256 rows (the raw reshape makes the C-pool contiguous). One block per
//     (b,l) group; WMMA tiles reduced into an LDS column accumulator.
// ---------------------------------------------------------------------------
__global__ __launch_bounds__(256) void k_proj_pool(const float* __restrict__ ctx,
                                                   const float* __restrict__ wout,
                                                   const float* __restrict__ bout,
                                                   float* __restrict__ pooled) {
  __shared__ float pool[EE];
  const int tid = threadIdx.x;
  if (tid < EE) pool[tid] = 0.0f;
  __syncthreads();

  const int lane = tid & 31;
  const int wid  = tid >> 5;
  const int r  = lane & 15;
  const int hf = lane >> 4;
  const size_t mbase = (size_t)blockIdx.x * 256;

  #pragma unroll
  for (int i = 0; i < 8; ++i) {
    const int t  = wid * 8 + i;          // 64 tiles: 16 row-tiles x 4 col-tiles
    const int m0 = (t >> 2) * 16;
    const int n0 = (t & 3) * 16;
    v8f acc = {0, 0, 0, 0, 0, 0, 0, 0};
    #pragma unroll
    for (int k0 = 0; k0 < 64; k0 += 4) {
      const int c0 = k0 + 2 * hf;
      const float* ap = ctx + (mbase + m0 + r) * EE + c0;
      v2f a; a.x = ap[0]; a.y = ap[1];
      const float* bp = wout + (size_t)c0 * EE + n0 + r;
      v2f b; b.x = bp[0]; b.y = bp[EE];
      acc = wmma4(a, b, acc);
    }
    float ps = acc[0] + acc[1] + acc[2] + acc[3] +
               acc[4] + acc[5] + acc[6] + acc[7];   // 8 rows, fixed col n0+r
    ps += __shfl_xor(ps, 16, 32);                    // + other 8 rows
    if (lane < 16) atomicAdd(&pool[n0 + r], ps);
  }
  __syncthreads();
  if (tid < EE)
    pooled[(size_t)blockIdx.x * EE + tid] = pool[tid] * (1.0f / 256.0f) + bout[tid];
}

// ---------------------------------------------------------------------------
// K6: MLP -> LN -> gelu -> MLP -> build L (unit lower) / U (softplus diag)
//     and M = L @ U per head. One block per (b,l,branch).
// ---------------------------------------------------------------------------
__global__ __launch_bounds__(256) void k_mlp_lu(
    const float* __restrict__ pooled,
    const float* __restrict__ mw1, const float* __restrict__ mb1,
    const float* __restrict__ mg,  const float* __restrict__ mbb,
    const float* __restrict__ mw2, const float* __restrict__ mb2,
    const float* __restrict__ nw1, const float* __restrict__ nb1,
    const float* __restrict__ ng,  const float* __restrict__ nbb,
    const float* __restrict__ nw2, const float* __restrict__ nb2,
    float* __restrict__ out) {
  const int branch = blockIdx.x & 1;
  const int row    = blockIdx.x >> 1;          // 0..383  (b*12 + l)
  const float* w1 = branch ? nw1 : mw1;
  const float* b1 = branch ? nb1 : mb1;
  const float* g  = branch ? ng  : mg;
  const float* bb = branch ? nbb : mbb;
  const float* w2 = branch ? nw2 : mw2;
  const float* b2 = branch ? nb2 : mb2;

  __shared__ float sx[EE];
  __shared__ float sh[HIDD];
  __shared__ float sraw[512];
  const int tid = threadIdx.x;

  if (tid < EE) sx[tid] = pooled[(size_t)row * EE + tid];
  __syncthreads();

  if (tid < HIDD) {
    float acc = b1[tid];
    #pragma unroll
    for (int i = 0; i < EE; ++i) acc += sx[i] * w1[i * HIDD + tid];
    float mu = acc;
    #pragma unroll
    for (int m = 16; m >= 1; m >>= 1) mu += __shfl_xor(mu, m, 32);
    mu *= (1.0f / 32.0f);
    const float d = acc - mu;
    float vv = d * d;
    #pragma unroll
    for (int m = 16; m >= 1; m >>= 1) vv += __shfl_xor(vv, m, 32);
    vv *= (1.0f / 32.0f);
    const float y = d * rsqrtf(vv + 1e-5f) * g[tid] + bb[tid];
    sh[tid] = 0.5f * y * (1.0f + erff(y * 0.70710678118654752f));  // exact gelu
  }
  __syncthreads();

  for (int c = tid; c < 512; c += 256) {
    float acc = b2[c];
    #pragma unroll
    for (int j = 0; j < HIDD; ++j) acc += sh[j] * w2[j * 512 + c];
    sraw[c] = acc;
  }
  __syncthreads();

  const size_t obase = ((size_t)branch * 384 + row) * 512;
  for (int o = tid; o < 512; o += 256) {
    const int h  = o >> 6;
    const int ik = o & 63;
    const int i  = ik >> 3;
    const int k  = ik & 7;
    const float* rh = sraw + h * 64;   // [0..27] strict-lower L, [28..63] upper U
    float acc = 0.0f;
    #pragma unroll
    for (int j = 0; j < NKK; ++j) {
      if (j > i || j > k) continue;
      const float Lij = (j == i) ? 1.0f : rh[(i * (i - 1)) / 2 + j];
      float u = rh[28 + (NKK * j - (j * (j - 1)) / 2) + (k - j)];
      if (j == k) u = softplusf(u) + 0.01f;
      acc += Lij * u;
    }
    out[obase + o] = acc;
  }
}

// ---------------------------------------------------------------------------
// launch
// ---------------------------------------------------------------------------
extern "C" void kernel_launch(void* const* d_in, const int* in_sizes, int n_in,
                              void* d_out, int out_size, void* d_ws, size_t ws_size,
                              hipStream_t stream) {
  const float* wsfeat = (const float*)d_in[0];
  const float* W_emb  = (const float*)d_in[1];
  const float* b_emb  = (const float*)d_in[2];
  const float* pos    = (const float*)d_in[3];
  const float* w_in   = (const float*)d_in[4];
  const float* b_in   = (const float*)d_in[5];
  const float* w_out  = (const float*)d_in[6];
  const float* b_out  = (const float*)d_in[7];
  const float* m_w1 = (const float*)d_in[8];
  const float* m_b1 = (const float*)d_in[9];
  const float* m_g  = (const float*)d_in[10];
  const float* m_bb = (const float*)d_in[11];
  const float* m_w2 = (const float*)d_in[12];
  const float* m_b2 = (const float*)d_in[13];
  const float* n_w1 = (const float*)d_in[14];
  const float* n_b1 = (const float*)d_in[15];
  const float* n_g  = (const float*)d_in[16];
  const float* n_bb = (const float*)d_in[17];
  const float* n_w2 = (const float*)d_in[18];
  const float* n_b2 = (const float*)d_in[19];
  float* out = (float*)d_out;

  float* ws     = (float*)d_ws;
  float* feats  = ws;                                   // 98304*56
  float* emb    = feats + (size_t)MROWS * 56;           // 98304*64
  float* qkv    = emb   + (size_t)MROWS * EE;           // 98304*192
  float* ctx    = qkv   + (size_t)MROWS * 192;          // 98304*64
  float* pooled = ctx   + (size_t)MROWS * EE;           // 384*64

  k_stats<<<SROWS / 8, 256, 0, stream>>>(wsfeat, feats);
  k_embed<<<MROWS / 32, 256, 0, stream>>>(feats, W_emb, b_emb, pos, emb);
  k_qkv<<<MROWS / 16, 384, 0, stream>>>(emb, w_in, b_in, qkv);
  k_attn<<<(BB * CC * NHH) / 8, 256, 0, stream>>>(qkv, ctx);
  k_proj_pool<<<BB * LL, 256, 0, stream>>>(ctx, w_out, b_out, pooled);
  k_mlp_lu<<<BB * LL * 2, 256, 0, stream>>>(pooled,
      m_w1, m_b1, m_g, m_bb, m_w2, m_b2,
      n_w1, n_b1, n_g, n_bb, n_w2, n_b2, out);
}